// GlobalPatternRegularizer_25812753449664
// MI455X (gfx1250) — compile-verified
//
#include <hip/hip_runtime.h>

#define NUM_GRAPHS   1024
#define NUM_ATOMS    64
#define REUSE_WEIGHT 0.01f

typedef float v2f __attribute__((ext_vector_type(2)));
typedef float v8f __attribute__((ext_vector_type(8)));

__device__ __forceinline__ void atomic_add_f32(float* p, float v) {
    __hip_atomic_fetch_add(p, v, __ATOMIC_RELAXED, __HIP_MEMORY_SCOPE_AGENT);
}

// ---------------------------------------------------------------------------
// Kernel 1: zero the workspace (seg sums + counts). Must run every call.
// ---------------------------------------------------------------------------
__global__ void zero_ws_kernel(float* __restrict__ p, int n) {
    int i = blockIdx.x * blockDim.x + threadIdx.x;
    if (i < n) p[i] = 0.0f;
}

// ---------------------------------------------------------------------------
// Kernel 2: streaming segment partial sums over the sorted batch vector.
// Each wave owns a contiguous run of rows; lane l accumulates columns
// (2l, 2l+1) in registers and flushes with f32 atomics only at segment
// boundaries. 512 MB streamed => HBM-bound (~22 us floor at 23.3 TB/s).
// ---------------------------------------------------------------------------
__global__ __launch_bounds__(256) void seg_partial_kernel(
    const float* __restrict__ codes, const int* __restrict__ batch,
    float* __restrict__ segsum, float* __restrict__ counts,
    int nrows, int rows_per_wave)
{
    const int gtid = blockIdx.x * blockDim.x + threadIdx.x;
    const int wave = gtid >> 5;
    const int lane = threadIdx.x & 31;

    long r0 = (long)wave * rows_per_wave;
    long r1 = r0 + rows_per_wave;
    if (r1 > nrows) r1 = nrows;
    if (r0 >= r1) return;  // uniform per wave

    const float2* __restrict__ codes2 = reinterpret_cast<const float2*>(codes);

    int   cur = batch[r0];
    float ax = 0.0f, ay = 0.0f;
    int   cnt = 0;

    for (long r = r0; r < r1; ++r) {
        // speculative lookahead prefetch (global_prefetch_b8); OOB is safe
        __builtin_prefetch(&codes2[(r + 24) * 32 + lane], 0, 0);
        int s = batch[r];
        if (s != cur) {
            float* dst = segsum + (long)cur * NUM_ATOMS + 2 * lane;
            atomic_add_f32(dst,     ax);
            atomic_add_f32(dst + 1, ay);
            if (lane == 0) atomic_add_f32(counts + cur, (float)cnt);
            ax = 0.0f; ay = 0.0f; cnt = 0; cur = s;
        }
        float2 v = codes2[r * 32 + lane];
        ax += v.x; ay += v.y; ++cnt;
    }
    float* dst = segsum + (long)cur * NUM_ATOMS + 2 * lane;
    atomic_add_f32(dst,     ax);
    atomic_add_f32(dst + 1, ay);
    if (lane == 0) atomic_add_f32(counts + cur, (float)cnt);
}

// ---------------------------------------------------------------------------
// Kernel 3: finalize. seg -> graph_codes (divide by counts, in place), then
// per-atom Sum and SumSq over 1024 graphs via V_WMMA_F32_16X16X4_F32:
//   SumSq = diag(G^T G) (16x16 Gram tile per 16-atom block)
//   Sum   = ones^T G    (any row of the D tile)
// For the f32 16x4 A / 4x16 B VGPR layouts the per-lane contents of A and B
// are identical: lane L holds {G[g0+2*(L>>4)][base+(L&15)], G[g0+1+2*(L>>4)][...]}.
// ---------------------------------------------------------------------------
__global__ __launch_bounds__(256) void finalize_kernel(
    float* __restrict__ seg, const float* __restrict__ counts,
    float* __restrict__ out)
{
    __shared__ float lq[4][8][32];   // SumSq D-tiles (per 16-atom block)
    __shared__ float lsums[4][32];   // VGPR0 row (M=0) of Sum D-tiles
    __shared__ float vbuf[NUM_ATOMS];

    const int tid = threadIdx.x;

    // Step 1: graph_codes = seg_sum / counts (in place)
    for (int idx = tid; idx < NUM_GRAPHS * NUM_ATOMS; idx += 256) {
        int g = idx >> 6;                  // NUM_ATOMS == 64
        float c = counts[g];
        float inv = (c > 0.0f) ? (1.0f / c) : 0.0f;
        seg[idx] *= inv;
    }
    __syncthreads();

    // Step 2: WMMA accumulation, one wave per 16-atom block (waves 0..3).
    const int wave = tid >> 5;
    const int lane = tid & 31;
    if (wave < 4) {
        const int base  = wave * 16;
        const int m     = lane & 15;
        const int khalf = lane >> 4;
        const int col   = base + m;

        v8f qacc = {};   // Gram-diagonal tile accumulator (SumSq)
        v8f sacc = {};   // ones^T G accumulator (Sum; rows identical)
        v2f ones; ones.x = 1.0f; ones.y = 1.0f;

        for (int g0 = 0; g0 < NUM_GRAPHS; g0 += 4) {
            const int r = g0 + 2 * khalf;
            v2f av;
            av.x = seg[r * NUM_ATOMS + col];
            av.y = seg[(r + 1) * NUM_ATOMS + col];
            qacc = __builtin_amdgcn_wmma_f32_16x16x4_f32(
                false, av, false, av, (short)0, qacc, false, false);
            sacc = __builtin_amdgcn_wmma_f32_16x16x4_f32(
                false, ones, false, av, (short)0, sacc, false, false);
        }
#pragma unroll
        for (int rr = 0; rr < 8; ++rr) lq[wave][rr][lane] = qacc[rr];
        lsums[wave][lane] = sacc[0];
    }
    __syncthreads();

    // Step 3: unbiased variance per atom, then mean and scale.
    if (tid < NUM_ATOMS) {
        const int w = tid >> 4;
        const int i = tid & 15;
        // D[i][i]: VGPR (i&7); lane i for i<8, lane i+16 for i>=8
        float Q = (i < 8) ? lq[w][i][i] : lq[w][i - 8][i + 16];
        float S = lsums[w][i];           // M=0 row, N=i -> VGPR0, lane i
        float var = (Q - S * S * (1.0f / NUM_GRAPHS)) * (1.0f / (NUM_GRAPHS - 1));
        vbuf[tid] = var;
    }
    __syncthreads();
    if (tid == 0) {
        float s = 0.0f;
        for (int a = 0; a < NUM_ATOMS; ++a) s += vbuf[a];
        out[0] = -REUSE_WEIGHT * (s * (1.0f / NUM_ATOMS));
    }
}

// ---------------------------------------------------------------------------
// Host launcher
// ---------------------------------------------------------------------------
extern "C" void kernel_launch(void* const* d_in, const int* in_sizes, int n_in,
                              void* d_out, int out_size, void* d_ws, size_t ws_size,
                              hipStream_t stream) {
    const float* codes = (const float*)d_in[0];   // [nrows, 64] fp32
    const int*   batch = (const int*)d_in[1];     // [nrows] int32, sorted
    const int    nrows = in_sizes[1];

    float* segsum = (float*)d_ws;                   // [1024][64]
    float* counts = segsum + NUM_GRAPHS * NUM_ATOMS; // [1024]
    float* out    = (float*)d_out;

    const int ztotal = NUM_GRAPHS * NUM_ATOMS + NUM_GRAPHS;
    zero_ws_kernel<<<(ztotal + 255) / 256, 256, 0, stream>>>(segsum, ztotal);

    const int waves  = 8192;                 // 1024 blocks x 8 waves (wave32)
    const int rpw    = (nrows + waves - 1) / waves;
    seg_partial_kernel<<<waves / 8, 256, 0, stream>>>(
        codes, batch, segsum, counts, nrows, rpw);

    finalize_kernel<<<1, 256, 0, stream>>>(segsum, counts, out);
}